// IncConvModule_38001870635792
// MI455X (gfx1250) — compile-verified
//
#include <hip/hip_runtime.h>
#include <hip/hip_bf16.h>

typedef __attribute__((ext_vector_type(2))) float v2f;
typedef __attribute__((ext_vector_type(8))) float v8f;

#define BATCH   64
#define CIN     64
#define COUT    64
#define Hh      112
#define Ww      112
#define HOUT    112
#define WOUT    112
#define OUT_PH  18
#define OUT_PW  18
#define M_TOT   (OUT_PH * OUT_PW)        // 324
#define M_TILES 21                       // ceil(324/16)
#define HW      (Hh * Ww)                // 12544
#define OUT_ELEMS ((size_t)BATCH * COUT * HOUT * WOUT)  // 51,380,224

// ---------------------------------------------------------------------------
// Kernel 1: bulk copy of out_tensor -> d_out (HBM-bound, ~411 MB total traffic)
// ---------------------------------------------------------------------------
__global__ __launch_bounds__(256) void copy_out_kernel(const float4* __restrict__ src,
                                                       float4* __restrict__ dst,
                                                       int n4) {
    int i = blockIdx.x * blockDim.x + threadIdx.x;
    int stride = gridDim.x * blockDim.x;
    for (; i < n4; i += stride) dst[i] = src[i];
}

// ---------------------------------------------------------------------------
// Kernel 2: write new_loc (int32 pairs) after the float tensor in d_out
// ---------------------------------------------------------------------------
__global__ void write_loc_kernel(const int* __restrict__ loc, int* __restrict__ newloc) {
    int b = threadIdx.x;
    if (b >= BATCH) return;
    int py = loc[2 * b + 0], px = loc[2 * b + 1];
    int oy = py - 1; oy = oy < 0 ? 0 : (oy > HOUT - OUT_PH ? HOUT - OUT_PH : oy);
    int ox = px - 1; ox = ox < 0 ? 0 : (ox > WOUT - OUT_PW ? WOUT - OUT_PW : ox);
    newloc[2 * b + 0] = oy;
    newloc[2 * b + 1] = ox;
}

// ---------------------------------------------------------------------------
// Kernel 3: 18x18 patch conv as implicit GEMM on fp32 WMMA (16x16x4).
// grid = (BATCH, M_TILES), block = 128 (4 waves, wave w = N-tile w).
// A[m][k]: patch pixels (m -> dy,dx ; k = tap*64 + cin), B[k][n]: weights.
// Padding handled branchlessly: clamp address in-range + multiply by 0/1 mask,
// so loads are unconditional, EXEC stays all-ones, no saveexec in the loop.
// ---------------------------------------------------------------------------
__global__ __launch_bounds__(128) void conv_patch_wmma_kernel(
    const float* __restrict__ x,     // [B, CIN, H, W]
    const float* __restrict__ wts,   // [COUT, CIN, 3, 3]
    const float* __restrict__ bias,  // [COUT]
    const int*   __restrict__ loc,   // [B, 2]
    float* __restrict__ out)         // [B, COUT, HOUT, WOUT]
{
    const int b     = blockIdx.x;
    const int mtile = blockIdx.y;
    const int ntile = threadIdx.x >> 5;   // wave id: 0..3
    const int lane  = threadIdx.x & 31;
    const int half  = lane >> 4;          // 0: lanes 0-15, 1: lanes 16-31
    const int lm    = lane & 15;

    int py = loc[2 * b + 0], px = loc[2 * b + 1];
    int oy = py - 1; oy = oy < 0 ? 0 : (oy > HOUT - OUT_PH ? HOUT - OUT_PH : oy);
    int ox = px - 1; ox = ox < 0 ? 0 : (ox > WOUT - OUT_PW ? WOUT - OUT_PW : ox);

    // A row this lane feeds (ISA layout: lanes 0-15 and 16-31 share M = lane%16)
    int m  = mtile * 16 + lm;
    int mc = m < M_TOT ? m : (M_TOT - 1);       // clamp: garbage rows discarded on store
    int dy0 = mc / OUT_PW;
    int dx0 = mc - dy0 * OUT_PW;

    const int n = ntile * 16 + lm;              // B/D column (cout) for this lane

    const float* xb = x + (size_t)b * CIN * HW + (size_t)(half * 2) * HW;
    const float* wb = wts + (size_t)n * CIN * 9 + (half * 2) * 9;

    v8f acc = {};

    for (int tap = 0; tap < 9; ++tap) {
        int ky = tap / 3, kx = tap - ky * 3;
        int y  = oy - 1 + dy0 + ky;             // global input row (pad=1)
        int xx = ox - 1 + dx0 + kx;             // global input col
        bool valid = ((unsigned)y < (unsigned)Hh) && ((unsigned)xx < (unsigned)Ww);
        float msk = valid ? 1.0f : 0.0f;        // branchless zero-padding
        int sp = valid ? (y * Ww + xx) : 0;     // clamped in-range address
        const float* xp = xb + sp;
        const float* wp = wb + tap;

        #pragma unroll
        for (int cg = 0; cg < 16; ++cg) {       // cin groups of 4 (K-step = 4)
            int c0 = cg * 4;
            v2f a, bm;
            a.x  = xp[(size_t)(c0)     * HW] * msk;
            a.y  = xp[(size_t)(c0 + 1) * HW] * msk;
            bm.x = wp[(c0)     * 9];
            bm.y = wp[(c0 + 1) * 9];
            // D = A(16x4 f32) * B(4x16 f32) + C   -> v_wmma_f32_16x16x4_f32
            acc = __builtin_amdgcn_wmma_f32_16x16x4_f32(
                false, a, false, bm, (short)0, acc, false, false);
        }
    }

    float bn = bias[n];
    #pragma unroll
    for (int r = 0; r < 8; ++r) {
        int mo = mtile * 16 + r + half * 8;     // D layout: VGPR r -> M = r + half*8
        if (mo < M_TOT) {
            int dy = mo / OUT_PW;
            int dx = mo - dy * OUT_PW;
            out[(((size_t)b * COUT + n) * HOUT + (oy + dy)) * WOUT + (ox + dx)] = acc[r] + bn;
        }
    }
}

// ---------------------------------------------------------------------------
extern "C" void kernel_launch(void* const* d_in, const int* in_sizes, int n_in,
                              void* d_out, int out_size, void* d_ws, size_t ws_size,
                              hipStream_t stream) {
    const float* x      = (const float*)d_in[0];   // in_tensor
    const float* wts    = (const float*)d_in[1];   // weights
    const float* bias   = (const float*)d_in[2];   // biases
    const float* out_in = (const float*)d_in[3];   // out_tensor
    const int*   loc    = (const int*)d_in[4];     // patch_location_tensor
    // d_in[5]/d_in[6] = p_height/p_width == 16 (fixed by setup; patch path always taken)

    float* out = (float*)d_out;

    // 1) Bulk copy out_tensor -> out (float4 = B128 loads/stores)
    int n4 = (int)(OUT_ELEMS / 4);                 // 12,845,056
    copy_out_kernel<<<6144, 256, 0, stream>>>((const float4*)out_in, (float4*)out, n4);

    // 2) new_loc (int32) appended after the float tensor
    write_loc_kernel<<<1, 64, 0, stream>>>(loc, (int*)(out + OUT_ELEMS));

    // 3) WMMA patch conv overwrites the 18x18 region (stream-ordered after copy)
    dim3 grid(BATCH, M_TILES);
    conv_patch_wmma_kernel<<<grid, 128, 0, stream>>>(x, wts, bias, loc, out);
}